// TensorProductLayer_23545010717445
// MI455X (gfx1250) — compile-verified
//
#include <hip/hip_runtime.h>

// Fused equivariant tensor-product message passing for MI455X (gfx1250).
//
// Strategy (bandwidth-bound problem, ~2 FLOP/byte):
//  - One pass over edges. Each wave32 owns a tile of 16 edges.
//  - The only GEMM-shaped piece (rbf [16x5(pad 8)] x coeffs^T [8x16] per path,
//    fp32) runs on V_WMMA_F32_16X16X4_F32: 2 K-chunks x 4 paths = 8 WMMAs/tile.
//  - The WMMA D layout (lane = channel, vgpr+half = edge) is exactly the
//    per-(edge,channel) layout the depthwise CG tensor product needs, so the
//    radial values never leave registers.
//  - segment_sum via global float atomics into the L2-resident output.

typedef __attribute__((ext_vector_type(2))) float v2f;
typedef __attribute__((ext_vector_type(8))) float v8f;

#define WMMA4F32(A, B, C) \
  __builtin_amdgcn_wmma_f32_16x16x4_f32(false, (A), false, (B), (short)0, (C), false, false)

__device__ __forceinline__ float rbf_eval(float r, float center) {
  float d = r - center;
  return __expf(-0.7f * d * d);  // spread = (3.5-0)/5 = 0.7
}

// ---------------------------------------------------------------------------
// Output init: coords pass-through + zero the accumulators (d_out is poisoned).
// ---------------------------------------------------------------------------
__global__ void init_out_kernel(const float* __restrict__ coords,
                                float* __restrict__ out,
                                int nCoordElems, int totalElems) {
  int idx = blockIdx.x * blockDim.x + threadIdx.x;
  if (idx < totalElems) {
    out[idx] = (idx < nCoordElems) ? coords[idx] : 0.0f;
  }
}

// ---------------------------------------------------------------------------
// Fused edge kernel. One wave per 16-edge tile.
// ---------------------------------------------------------------------------
__global__ void __launch_bounds__(256)
edge_tp_kernel(const float* __restrict__ nodes0,    // [N,16]
               const float* __restrict__ nodes1,    // [N,16,3]
               const float* __restrict__ edges,     // [E,3]
               const int*   __restrict__ senders,   // [E]
               const int*   __restrict__ receivers, // [E]
               const float* __restrict__ c00, const float* __restrict__ c01,
               const float* __restrict__ c10, const float* __restrict__ c11,
               const float* __restrict__ w000, const float* __restrict__ w011,
               const float* __restrict__ w101, const float* __restrict__ w110,
               const float* __restrict__ w111,
               float* __restrict__ out0,            // [N,32]
               float* __restrict__ out1,            // [N,48,3]
               int E, int nTiles) {
  const int lane = threadIdx.x & 31;
  const int wave = threadIdx.x >> 5;
  const int half = lane >> 4;   // 0: lanes 0-15, 1: lanes 16-31
  const int c    = lane & 15;   // channel (N index) == edge-in-tile (M index)
  const int kb   = half * 2;    // K offset inside a 4-wide K chunk

  const int tile = blockIdx.x * (blockDim.x >> 5) + wave;
  if (tile >= nTiles) return;   // uniform per wave -> EXEC stays all-ones

  // --- B fragments: coeffs^T, K (basis, padded to 8) x N (16 channels). ---
  // Layout mirror of A: vgpr0 holds row K=kb, vgpr1 holds row K=kb+1.
  v2f b0_00, b0_01, b0_10, b0_11;   // K chunk 0 (rows kb, kb+1)
  v2f b1_00, b1_01, b1_10, b1_11;   // K chunk 1 (rows 4+kb, 5+kb; >=5 -> 0)
  {
    const int o = c * 5 + kb;       // coeffs row-major [16,5]
    b0_00.x = c00[o]; b0_00.y = c00[o + 1];
    b0_01.x = c01[o]; b0_01.y = c01[o + 1];
    b0_10.x = c10[o]; b0_10.y = c10[o + 1];
    b0_11.x = c11[o]; b0_11.y = c11[o + 1];
    const bool low = (kb == 0);     // row 4 lives in lanes 0-15, vgpr0
    b1_00.x = low ? c00[c * 5 + 4] : 0.f; b1_00.y = 0.f;
    b1_01.x = low ? c01[c * 5 + 4] : 0.f; b1_01.y = 0.f;
    b1_10.x = low ? c10[c * 5 + 4] : 0.f; b1_10.y = 0.f;
    b1_11.x = low ? c11[c * 5 + 4] : 0.f; b1_11.y = 0.f;
  }

  // --- Per-channel path weights (CG norms folded in). ---
  const float W000 = w000[c];
  const float W011 = w011[c];
  const float W101 = w101[c];
  const float W110 = w110[c] * 0.57735026918962576f;  // 1/sqrt(3)
  const float W111 = w111[c] * 0.70710678118654752f;  // 1/sqrt(2)

  const int base = tile * 16;

  // --- A fragments: 16 edges x K. Lane (half h) holds edge M=c, rows kb,kb+1.
  v2f a0, a1;
  {
    int eA = base + c;
    int eC = (eA < E) ? eA : (E - 1);   // clamp; dead rows never read back
    float ex = edges[eC * 3 + 0];
    float ey = edges[eC * 3 + 1];
    float ez = edges[eC * 3 + 2];
    float r = sqrtf(ex * ex + ey * ey + ez * ez);
    a0.x = rbf_eval(r, 0.875f * (float)kb);
    a0.y = rbf_eval(r, 0.875f * (float)(kb + 1));
    a1.x = (kb == 0) ? rbf_eval(r, 3.5f) : 0.f;   // basis k=4; k>=5 padded 0
    a1.y = 0.f;
  }

  // --- rad_p = rbfs @ coeffs_p^T via WMMA (fp32, exact). ---
  v8f r00 = {}; v8f r01 = {}; v8f r10 = {}; v8f r11 = {};
  r00 = WMMA4F32(a0, b0_00, r00);
  r01 = WMMA4F32(a0, b0_01, r01);
  r10 = WMMA4F32(a0, b0_10, r10);
  r11 = WMMA4F32(a0, b0_11, r11);
  r00 = WMMA4F32(a1, b1_00, r00);
  r01 = WMMA4F32(a1, b1_01, r01);
  r10 = WMMA4F32(a1, b1_10, r10);
  r11 = WMMA4F32(a1, b1_11, r11);

  // --- Depthwise CG tensor product + scatter. ---
  // D layout: vgpr v, this lane -> (edge M = v + 8*half, channel N = c).
#pragma unroll
  for (int v = 0; v < 8; ++v) {
    const int e = base + v + half * 8;
    if (e < E) {
      const float ex = edges[e * 3 + 0];
      const float ey = edges[e * 3 + 1];
      const float ez = edges[e * 3 + 2];
      const int s = senders[e];
      const int rc = receivers[e];

      const float n0 = nodes0[s * 16 + c];
      const float* n1p = nodes1 + (s * 16 + c) * 3;
      const float n1x = n1p[0], n1y = n1p[1], n1z = n1p[2];

      const float rad00 = r00[v], rad01 = r01[v], rad10 = r10[v], rad11 = r11[v];

      const float f00 = rad00 * 0.28209479177387814f;   // Y00
      const float f10 = rad10 * 0.28209479177387814f;
      // solid harmonics l=1, real basis order (y, z, x)
      const float yx = 0.4886025119029199f * ey;
      const float yy = 0.4886025119029199f * ez;
      const float yz = 0.4886025119029199f * ex;
      const float f01x = rad01 * yx, f01y = rad01 * yy, f01z = rad01 * yz;
      const float f11x = rad11 * yx, f11y = rad11 * yy, f11z = rad11 * yz;

      // l_o = 0: path (0,0) identity; path (1,1) dot/sqrt(3)
      atomicAdd(&out0[rc * 32 + c],      W000 * n0 * f00);
      atomicAdd(&out0[rc * 32 + 16 + c],
                W110 * (n1x * f11x + n1y * f11y + n1z * f11z));

      // l_o = 1: (0,1) n0*f01 ; (1,0) f10*n1 ; (1,1) cross/sqrt(2)
      float* o1 = out1 + rc * 144;
      atomicAdd(&o1[c * 3 + 0], W011 * n0 * f01x);
      atomicAdd(&o1[c * 3 + 1], W011 * n0 * f01y);
      atomicAdd(&o1[c * 3 + 2], W011 * n0 * f01z);

      atomicAdd(&o1[(16 + c) * 3 + 0], W101 * f10 * n1x);
      atomicAdd(&o1[(16 + c) * 3 + 1], W101 * f10 * n1y);
      atomicAdd(&o1[(16 + c) * 3 + 2], W101 * f10 * n1z);

      const float cx = n1y * f11z - n1z * f11y;
      const float cy = n1z * f11x - n1x * f11z;
      const float cz = n1x * f11y - n1y * f11x;
      atomicAdd(&o1[(32 + c) * 3 + 0], W111 * cx);
      atomicAdd(&o1[(32 + c) * 3 + 1], W111 * cy);
      atomicAdd(&o1[(32 + c) * 3 + 2], W111 * cz);
    }
  }
}

// ---------------------------------------------------------------------------
extern "C" void kernel_launch(void* const* d_in, const int* in_sizes, int n_in,
                              void* d_out, int out_size, void* d_ws, size_t ws_size,
                              hipStream_t stream) {
  const float* coords    = (const float*)d_in[0];
  const float* nodes0    = (const float*)d_in[1];
  const float* nodes1    = (const float*)d_in[2];
  const float* edges     = (const float*)d_in[3];
  const int*   senders   = (const int*)d_in[4];
  const int*   receivers = (const int*)d_in[5];
  const float* c00  = (const float*)d_in[6];
  const float* c01  = (const float*)d_in[7];
  const float* c10  = (const float*)d_in[8];
  const float* c11  = (const float*)d_in[9];
  const float* w000 = (const float*)d_in[10];
  const float* w011 = (const float*)d_in[11];
  const float* w101 = (const float*)d_in[12];
  const float* w110 = (const float*)d_in[13];
  const float* w111 = (const float*)d_in[14];

  const int E = in_sizes[4];            // senders count
  const int nNodes = in_sizes[0] / 3;   // coords [N,3]

  float* out = (float*)d_out;
  float* out0 = out + (size_t)nNodes * 3;                     // [N,32]
  float* out1 = out0 + (size_t)nNodes * 32;                   // [N,48,3]
  const int total = nNodes * 3 + nNodes * 32 + nNodes * 144;

  init_out_kernel<<<(total + 255) / 256, 256, 0, stream>>>(coords, out,
                                                           nNodes * 3, total);

  const int nTiles = (E + 15) / 16;
  const int wavesPerBlock = 256 / 32;
  const int blocks = (nTiles + wavesPerBlock - 1) / wavesPerBlock;
  edge_tp_kernel<<<blocks, 256, 0, stream>>>(
      nodes0, nodes1, edges, senders, receivers,
      c00, c01, c10, c11, w000, w011, w101, w110, w111,
      out0, out1, E, nTiles);
}